// NGCF_74723841016248
// MI455X (gfx1250) — compile-verified
//
#include <hip/hip_runtime.h>
#include <hip/hip_bf16.h>
#include <math.h>

#define N_USERS 50000
#define N_ITEMS 100000
#define N_NODES 150000
#define N_EDGES 2000000
#define EMBED   64
#define N_LAYERS 3
#define BATCH   8192
#define NEG_SLOPE 0.01f
#define EPS 1e-12f
#define FIN_W (EMBED * (N_LAYERS + 1))   // 256

typedef __attribute__((ext_vector_type(2))) float v2f;
typedef __attribute__((ext_vector_type(8))) float v8f;

// ---------------------------------------------------------------------------
// emb = concat(user_emb, item_emb); fin[:, 0:64] = emb
// ---------------------------------------------------------------------------
__global__ void ngcf_init_kernel(const float* __restrict__ user_emb,
                                 const float* __restrict__ item_emb,
                                 float* __restrict__ emb,
                                 float* __restrict__ fin) {
  long long t = (long long)blockIdx.x * blockDim.x + threadIdx.x;
  if (t >= (long long)N_NODES * EMBED) return;
  int node = (int)(t >> 6);
  int c    = (int)(t & 63);
  float v = (node < N_USERS) ? user_emb[(long long)node * EMBED + c]
                             : item_emb[(long long)(node - N_USERS) * EMBED + c];
  emb[t] = v;
  fin[(long long)node * FIN_W + c] = v;
}

// ---------------------------------------------------------------------------
// SpMM: t1e[rows[e], :] += vals[e] * emb[cols[e], :]
// One wave32 per edge; each lane handles components {lane, lane+32} -> two
// fully coalesced 128B atomic bursts per edge. Table (38.4 MB) is L2 resident.
// ---------------------------------------------------------------------------
__global__ void ngcf_spmm_kernel(const int*   __restrict__ rows,
                                 const int*   __restrict__ cols,
                                 const float* __restrict__ vals,
                                 const float* __restrict__ x,
                                 float*       __restrict__ y) {
  long long gid = (long long)blockIdx.x * blockDim.x + threadIdx.x;
  int e    = (int)(gid >> 5);
  int lane = (int)(gid & 31);
  if (e >= N_EDGES) return;
  int r = rows[e];
  int c = cols[e];
  float v = vals[e];
  const float* src = x + (long long)c * EMBED;
  float*       dst = y + (long long)r * EMBED;
  atomicAdd(&dst[lane],      v * src[lane]);
  atomicAdd(&dst[lane + 32], v * src[lane + 32]);
}

// ---------------------------------------------------------------------------
// Fused dense layer:
//   out = leaky_relu( t1e @ W1^T + (emb .* t1e) @ W2^T + b1 + b2 )
// Block = 128 threads (4 waves). Block owns a 16-row tile; each wave owns one
// 16-column tile and accumulates K=64 with 16x2 V_WMMA_F32_16X16X4_F32 ops.
//
// f32 WMMA operand layouts (ISA 7.12.2):
//   A 16x4 : lane L, VGPR j -> A[L%16][k0 + 2*(L/16) + j]
//   B 4x16 : lane L, VGPR j -> B[k0 + 2*(L/16) + j][L%16]   (B[k][n] = W[n][k])
//   C 16x16: lane L, VGPR r -> C[r + 8*(L/16)][L%16]
// ---------------------------------------------------------------------------
__global__ void __launch_bounds__(128)
ngcf_gemm_layer_kernel(const float* __restrict__ t1e,
                       const float* __restrict__ emb,
                       const float* __restrict__ W1,
                       const float* __restrict__ b1,
                       const float* __restrict__ W2,
                       const float* __restrict__ b2,
                       float* __restrict__ out) {
  int rowbase = blockIdx.x * 16;
  int wave = threadIdx.x >> 5;         // n-tile 0..3
  int lane = threadIdx.x & 31;
  int lm   = lane & 15;
  int hi   = lane >> 4;
  int n    = wave * 16 + lm;           // output column for this lane

  // All 8 C values in this lane share column n -> seed accumulator with bias.
  float bb = b1[n] + b2[n];
  v8f acc;
#pragma unroll
  for (int r = 0; r < 8; ++r) acc[r] = bb;

  const float* t1row = t1e + (long long)(rowbase + lm) * EMBED;
  const float* erow  = emb + (long long)(rowbase + lm) * EMBED;
  const float* w1row = W1 + (long long)n * EMBED;
  const float* w2row = W2 + (long long)n * EMBED;

#pragma unroll
  for (int kk = 0; kk < 16; ++kk) {
    int k = kk * 4 + 2 * hi;
    v2f a1  = *(const v2f*)(t1row + k);          // A  = t1e tile
    v2f e1  = *(const v2f*)(erow  + k);
    v2f a2  = a1 * e1;                           // A' = emb .* t1e
    v2f bw1 = *(const v2f*)(w1row + k);          // B  = W1^T slice
    v2f bw2 = *(const v2f*)(w2row + k);          // B' = W2^T slice
    acc = __builtin_amdgcn_wmma_f32_16x16x4_f32(false, a1, false, bw1,
                                                (short)0, acc, false, false);
    acc = __builtin_amdgcn_wmma_f32_16x16x4_f32(false, a2, false, bw2,
                                                (short)0, acc, false, false);
  }

#pragma unroll
  for (int r = 0; r < 8; ++r) {
    int m = rowbase + r + 8 * hi;
    float x = acc[r];
    x = (x >= 0.0f) ? x : NEG_SLOPE * x;
    out[(long long)m * EMBED + n] = x;
  }
}

// ---------------------------------------------------------------------------
// Row L2-normalize into fin[:, col_off:col_off+64]. One wave32 per row.
// ---------------------------------------------------------------------------
__global__ void ngcf_norm_kernel(const float* __restrict__ emb,
                                 float* __restrict__ fin,
                                 int col_off) {
  long long gid = (long long)blockIdx.x * blockDim.x + threadIdx.x;
  int row  = (int)(gid >> 5);
  int lane = (int)(gid & 31);
  if (row >= N_NODES) return;
  const float* e = emb + (long long)row * EMBED;
  float x0 = e[lane];
  float x1 = e[lane + 32];
  float s = x0 * x0 + x1 * x1;
#pragma unroll
  for (int off = 16; off > 0; off >>= 1) s += __shfl_xor(s, off, 32);
  float inv = 1.0f / fmaxf(sqrtf(s), EPS);
  float* f = fin + (long long)row * FIN_W + col_off;
  f[lane]      = x0 * inv;
  f[lane + 32] = x1 * inv;
}

// ---------------------------------------------------------------------------
// BPR loss: one wave32 per batch element over 256-dim final embeddings.
// ---------------------------------------------------------------------------
__global__ void ngcf_loss_kernel(const float* __restrict__ fin,
                                 const int* __restrict__ u,
                                 const int* __restrict__ ii,
                                 const int* __restrict__ jj,
                                 float* __restrict__ out) {
  long long gid = (long long)blockIdx.x * blockDim.x + threadIdx.x;
  int b    = (int)(gid >> 5);
  int lane = (int)(gid & 31);
  if (b >= BATCH) return;
  const float* bu = fin + (long long)u[b] * FIN_W;
  const float* bp = fin + (long long)(N_USERS + ii[b]) * FIN_W;
  const float* bn = fin + (long long)(N_USERS + jj[b]) * FIN_W;
  float pos = 0.0f, neg = 0.0f;
#pragma unroll
  for (int t = 0; t < 8; ++t) {
    int c = lane + 32 * t;
    float a = bu[c];
    pos += a * bp[c];
    neg += a * bn[c];
  }
#pragma unroll
  for (int off = 16; off > 0; off >>= 1) {
    pos += __shfl_xor(pos, off, 32);
    neg += __shfl_xor(neg, off, 32);
  }
  if (lane == 0) {
    float x = pos - neg;
    // -log_sigmoid(x) = softplus(-x), numerically stable form
    float l = fmaxf(-x, 0.0f) + log1pf(expf(-fabsf(x)));
    atomicAdd(out, l * (1.0f / (float)BATCH));
  }
}

// ---------------------------------------------------------------------------
extern "C" void kernel_launch(void* const* d_in, const int* in_sizes, int n_in,
                              void* d_out, int out_size, void* d_ws, size_t ws_size,
                              hipStream_t stream) {
  (void)in_sizes; (void)n_in; (void)out_size; (void)ws_size;
  const float* user_emb = (const float*)d_in[0];
  const float* item_emb = (const float*)d_in[1];
  const float* W1       = (const float*)d_in[2];
  const float* b1       = (const float*)d_in[3];
  const float* W2       = (const float*)d_in[4];
  const float* b2       = (const float*)d_in[5];
  const float* adj_vals = (const float*)d_in[6];
  const int*   adj_rows = (const int*)d_in[7];
  const int*   adj_cols = (const int*)d_in[8];
  const int*   u_idx    = (const int*)d_in[9];
  const int*   i_idx    = (const int*)d_in[10];
  const int*   j_idx    = (const int*)d_in[11];
  float* out = (float*)d_out;

  char* ws = (char*)d_ws;
  size_t nodeEmbBytes = (size_t)N_NODES * EMBED * sizeof(float);
  float* emb  = (float*)(ws);
  float* t1e  = (float*)(ws + nodeEmbBytes);
  float* emb2 = (float*)(ws + 2 * nodeEmbBytes);
  float* fin  = (float*)(ws + 3 * nodeEmbBytes);   // N_NODES x 256

  {
    long long tot = (long long)N_NODES * EMBED;
    ngcf_init_kernel<<<(unsigned)((tot + 255) / 256), 256, 0, stream>>>(
        user_emb, item_emb, emb, fin);
  }

  float* cur = emb;
  float* nxt = emb2;
  for (int l = 0; l < N_LAYERS; ++l) {
    hipMemsetAsync(t1e, 0, nodeEmbBytes, stream);
    {
      long long tot = (long long)N_EDGES * 32;
      ngcf_spmm_kernel<<<(unsigned)((tot + 255) / 256), 256, 0, stream>>>(
          adj_rows, adj_cols, adj_vals, cur, t1e);
    }
    ngcf_gemm_layer_kernel<<<N_NODES / 16, 128, 0, stream>>>(
        t1e, cur,
        W1 + (size_t)l * EMBED * EMBED, b1 + (size_t)l * EMBED,
        W2 + (size_t)l * EMBED * EMBED, b2 + (size_t)l * EMBED,
        nxt);
    {
      long long tot = (long long)N_NODES * 32;
      ngcf_norm_kernel<<<(unsigned)((tot + 255) / 256), 256, 0, stream>>>(
          nxt, fin, (l + 1) * EMBED);
    }
    float* tmp = cur; cur = nxt; nxt = tmp;
  }

  hipMemsetAsync(d_out, 0, sizeof(float), stream);
  {
    long long tot = (long long)BATCH * 32;
    ngcf_loss_kernel<<<(unsigned)((tot + 255) / 256), 256, 0, stream>>>(
        fin, u_idx, i_idx, j_idx, out);
  }
}